// NEDT_68547678044278
// MI455X (gfx1250) — compile-verified
//
#include <hip/hip_runtime.h>
#include <hip/hip_bf16.h>

// ---------------------------------------------------------------------------
// NEDT pipeline for [16,3,540,960] f32 on gfx1250 (MI455X).
//  K1: gray + horizontal blur (both sigmas) via V_WMMA_F32_16X16X4_F32
//  K2: vertical blur via WMMA; tiles staged with GLOBAL_LOAD_ASYNC_TO_LDS_B32
//  K3: row EDT pass (binary input => two O(w) sweeps, exact) + prefetch
//  K4: column EDT pass (brute-force min-plus, async-LDS staged) + finalize
//
// Banded kernel matrices are served from zero-padded LDS tables so each B/A
// weight fetch is one unconditional ds_load (no EXEC-predicated global loads).
// ---------------------------------------------------------------------------

#define IMG_H 540
#define IMG_W 960
#define NBATCH 16
#define TILES_X (IMG_W / 16)            // 60
#define TILES_Y ((IMG_H + 15) / 16)     // 34
#define KEXT 28                         // 16 + 2*6 halo (largest pad = 6)
#define KSTEPS (KEXT / 4)               // 7 wmma k-steps
#define DIAM2 1213200.0f                // 540^2 + 960^2
#define WTAB 48                         // extended zero-padded weight table
#define WOFF 17                         // table offset: idx -17..+27 -> 0..44

typedef __attribute__((ext_vector_type(2))) float v2f;
typedef __attribute__((ext_vector_type(8))) float v8f;

// Normalized gaussian weights (compile-time; sigma0=1.0 ks=9, sigma1=1.6 ks=13)
__constant__ float W0C[9] = {
    1.3383086e-4f, 4.4318620e-3f, 5.3991130e-2f, 2.4197144e-1f, 3.9894348e-1f,
    2.4197144e-1f, 5.3991130e-2f, 4.4318620e-3f, 1.3383086e-4f};
__constant__ float W1C[13] = {
    2.2038e-4f, 1.88902e-3f, 1.095561e-2f, 4.299310e-2f, 1.1415989e-1f,
    2.0510803e-1f, 2.4934817e-1f, 2.0510803e-1f, 1.1415989e-1f, 4.299310e-2f,
    1.095561e-2f, 1.88902e-3f, 2.2038e-4f};

// Async global->LDS copy of one dword (CDNA5 path, ASYNCcnt-tracked, GVS mode).
__device__ __forceinline__ void async_g2l_b32(unsigned lds_addr, unsigned goff,
                                              unsigned long long gbase) {
    asm volatile("global_load_async_to_lds_b32 %0, %1, %2"
                 :: "v"(lds_addr), "v"(goff), "s"(gbase)
                 : "memory");
}
__device__ __forceinline__ void wait_async0() {
    asm volatile("s_wait_asynccnt 0" ::: "memory");
}

// Build zero-padded band tables in LDS (one wave per block).
__device__ __forceinline__ void build_weight_tables(float* w0e, float* w1e,
                                                    int lane) {
#pragma unroll
    for (int i = lane; i < WTAB; i += 32) {
        w0e[i] = 0.0f;
        w1e[i] = 0.0f;
    }
    __syncthreads();
    if (lane < 9)  w0e[WOFF + lane] = W0C[lane];
    if (lane < 13) w1e[WOFF + lane] = W1C[lane];
    __syncthreads();
}

// ---------------------------------------------------------------------------
// K1: gray + horizontal blur.  One wave per 16x16 output tile.
// out_tile[16x16] = E[16x28] (A) x Kmat[28x16] (B), chained 16x16x4 WMMAs.
// Kmat0[j,c] = w0[j-c-2] (pad 4, halo offset 6); Kmat1[j,c] = w1[j-c].
// ---------------------------------------------------------------------------
__global__ __launch_bounds__(32) void nedt_hblur(const float* __restrict__ img,
                                                 float* __restrict__ h0,
                                                 float* __restrict__ h1) {
    const int tx = blockIdx.x, ty = blockIdx.y, bi = blockIdx.z;
    const int lane = threadIdx.x;
    __shared__ float E[16][KEXT];
    __shared__ float w0e[WTAB];
    __shared__ float w1e[WTAB];

    build_weight_tables(w0e, w1e, lane);

    const float* rp = img + (size_t)bi * 3 * IMG_H * IMG_W;
    const float* gp = rp + (size_t)IMG_H * IMG_W;
    const float* bp = rp + (size_t)2 * IMG_H * IMG_W;

    const int x0 = tx * 16 - 6;  // halo left by 6
    const int y0 = ty * 16;

    // 16*28 = 448 elems, 14 per lane; replicate-clamp both axes.
#pragma unroll
    for (int i = lane; i < 16 * KEXT; i += 32) {
        int r = i / KEXT, c = i - r * KEXT;
        int y = min(y0 + r, IMG_H - 1);
        int x = min(max(x0 + c, 0), IMG_W - 1);
        size_t o = (size_t)y * IMG_W + x;
        E[r][c] = 0.299f * rp[o] + 0.587f * gp[o] + 0.114f * bp[o];
    }
    __syncthreads();

    const int m  = lane & 15;   // A row / B col / D col
    const int hf = lane >> 4;   // which K pair within a 4-step

    v8f acc0 = (v8f)(0.0f);
    v8f acc1 = (v8f)(0.0f);
#pragma unroll
    for (int ks = 0; ks < KSTEPS; ++ks) {
        const int k0 = ks * 4 + 2 * hf;
        v2f a = {E[m][k0], E[m][k0 + 1]};
        // banded constant B from zero-padded LDS tables (no branches)
        v2f b0 = {w0e[WOFF + k0 - m - 2], w0e[WOFF + k0 - m - 1]};
        v2f b1 = {w1e[WOFF + k0 - m],     w1e[WOFF + k0 - m + 1]};
        acc0 = __builtin_amdgcn_wmma_f32_16x16x4_f32(false, a, false, b0,
                                                     (short)0, acc0, false, false);
        acc1 = __builtin_amdgcn_wmma_f32_16x16x4_f32(false, a, false, b1,
                                                     (short)0, acc1, false, false);
    }

    const int xs = tx * 16 + m;
#pragma unroll
    for (int i = 0; i < 8; ++i) {
        int y = y0 + hf * 8 + i;
        if (y < IMG_H) {
            size_t o = (size_t)bi * IMG_H * IMG_W + (size_t)y * IMG_W + xs;
            h0[o] = acc0[i];
            h1[o] = acc1[i];
        }
    }
}

// ---------------------------------------------------------------------------
// K2: vertical blur + DoG threshold.  out = Kmat[16x28] (A) x E[28x16] (B).
// Tiles staged cache->LDS via GLOBAL_LOAD_ASYNC_TO_LDS_B32 (no VGPR bounce).
// data = (g1 - g0 > 0.005) ? 0 : diam2
// ---------------------------------------------------------------------------
__global__ __launch_bounds__(32) void nedt_vblur(const float* __restrict__ h0,
                                                 const float* __restrict__ h1,
                                                 float* __restrict__ data) {
    const int tx = blockIdx.x, ty = blockIdx.y, bi = blockIdx.z;
    const int lane = threadIdx.x;
    __shared__ float E0[KEXT][16];
    __shared__ float E1[KEXT][16];
    __shared__ float w0e[WTAB];
    __shared__ float w1e[WTAB];

    build_weight_tables(w0e, w1e, lane);

    const int x0 = tx * 16;
    const int y0 = ty * 16;
    const size_t base = (size_t)bi * IMG_H * IMG_W;

    // Uniform SGPR base addresses for this batch image.
    const unsigned long long h0b = (unsigned long long)(uintptr_t)(h0 + base);
    const unsigned long long h1b = (unsigned long long)(uintptr_t)(h1 + base);

    // 28*16 = 448 elems, 14 per lane; direct async copy into LDS.
#pragma unroll
    for (int i = lane; i < KEXT * 16; i += 32) {
        int r = i >> 4, c = i & 15;
        int y = min(max(y0 - 6 + r, 0), IMG_H - 1);
        unsigned goff = (unsigned)(((size_t)y * IMG_W + (x0 + c)) * sizeof(float));
        async_g2l_b32((unsigned)(uintptr_t)&E0[r][c], goff, h0b);
        async_g2l_b32((unsigned)(uintptr_t)&E1[r][c], goff, h1b);
    }
    wait_async0();
    __syncthreads();

    const int m  = lane & 15;
    const int hf = lane >> 4;

    v8f acc0 = (v8f)(0.0f);
    v8f acc1 = (v8f)(0.0f);
#pragma unroll
    for (int ks = 0; ks < KSTEPS; ++ks) {
        const int k0 = ks * 4 + 2 * hf;
        // A = banded constant kernel matrix: A[r,j] = w[j - r - off]
        v2f a0 = {w0e[WOFF + k0 - m - 2], w0e[WOFF + k0 - m - 1]};
        v2f a1 = {w1e[WOFF + k0 - m],     w1e[WOFF + k0 - m + 1]};
        v2f b0 = {E0[k0][m], E0[k0 + 1][m]};
        v2f b1 = {E1[k0][m], E1[k0 + 1][m]};
        acc0 = __builtin_amdgcn_wmma_f32_16x16x4_f32(false, a0, false, b0,
                                                     (short)0, acc0, false, false);
        acc1 = __builtin_amdgcn_wmma_f32_16x16x4_f32(false, a1, false, b1,
                                                     (short)0, acc1, false, false);
    }

    const int xs = x0 + m;
#pragma unroll
    for (int i = 0; i < 8; ++i) {
        int y = y0 + hf * 8 + i;
        if (y < IMG_H) {
            float diff = acc1[i] - acc0[i];        // g1 - g0
            float v = (diff > 0.005f) ? 0.0f : DIAM2;
            data[base + (size_t)y * IMG_W + xs] = v;
        }
    }
}

// ---------------------------------------------------------------------------
// K3: row EDT pass.  data is {0, diam2} so exact result = min(diam2, dmin^2)
// with dmin = distance to nearest zero; two linear sweeps per row.
// One thread per row (16*540 rows); global_prefetch ahead of the walk.
// ---------------------------------------------------------------------------
__global__ __launch_bounds__(256) void nedt_rowpass(const float* __restrict__ data,
                                                    float* __restrict__ inter) {
    int row = blockIdx.x * blockDim.x + threadIdx.x;
    if (row >= NBATCH * IMG_H) return;
    const float* d = data + (size_t)row * IMG_W;
    float* o = inter + (size_t)row * IMG_W;

    float dist = 2048.0f;  // 2048^2 > diam2 => clamps to diam2 when no zero
    for (int xc = 0; xc < IMG_W; xc += 16) {
        __builtin_prefetch(d + xc + 128, 0, 0);  // global_prefetch_b8, 2 lines ahead
#pragma unroll
        for (int x = xc; x < xc + 16; ++x) {
            dist = (d[x] == 0.0f) ? 0.0f : dist + 1.0f;
            o[x] = dist;  // left distance, combined in backward sweep
        }
    }
    dist = 2048.0f;
    for (int xc = IMG_W - 16; xc >= 0; xc -= 16) {
#pragma unroll
        for (int x = xc + 15; x >= xc; --x) {
            dist = (d[x] == 0.0f) ? 0.0f : dist + 1.0f;
            float dd = fminf(o[x], dist);
            o[x] = fminf(DIAM2, dd * dd);
        }
    }
}

// ---------------------------------------------------------------------------
// K4: column EDT pass (faithful brute force min-plus over h=540) + finalize.
// One block per (column, batch); column async-staged into LDS; in-place out.
// ---------------------------------------------------------------------------
__global__ __launch_bounds__(256) void nedt_colpass(float* __restrict__ out) {
    const int x = blockIdx.x;
    const int bi = blockIdx.y;
    __shared__ float col[IMG_H];
    float* base = out + (size_t)bi * IMG_H * IMG_W + x;
    const unsigned long long b64 = (unsigned long long)(uintptr_t)base;

    for (int r = threadIdx.x; r < IMG_H; r += 256) {
        unsigned goff = (unsigned)((size_t)r * IMG_W * sizeof(float));
        async_g2l_b32((unsigned)(uintptr_t)&col[r], goff, b64);
    }
    wait_async0();      // each wave drains its own ASYNCcnt ...
    __syncthreads();    // ... then the block barrier publishes LDS to all waves

    for (int r = threadIdx.x; r < IMG_H; r += 256) {
        float m = DIAM2;
        float fr = (float)r;
#pragma unroll 4
        for (int j = 0; j < IMG_H; ++j) {
            float dj = fr - (float)j;
            m = fminf(m, fmaf(dj, dj, col[j]));
        }
        float e = sqrtf(m);
        base[(size_t)r * IMG_W] = 1.0f - expf(-e * (36.0f / 960.0f));
    }
}

// ---------------------------------------------------------------------------
extern "C" void kernel_launch(void* const* d_in, const int* in_sizes, int n_in,
                              void* d_out, int out_size, void* d_ws, size_t ws_size,
                              hipStream_t stream) {
    (void)in_sizes; (void)n_in; (void)out_size; (void)ws_size;
    const float* img = (const float*)d_in[0];
    float* outp = (float*)d_out;

    const size_t NPIX = (size_t)NBATCH * IMG_H * IMG_W;  // 8,294,400
    float* h0   = (float*)d_ws;
    float* h1   = h0 + NPIX;
    float* dat  = h1 + NPIX;

    dim3 tiles(TILES_X, TILES_Y, NBATCH);
    nedt_hblur<<<tiles, 32, 0, stream>>>(img, h0, h1);
    nedt_vblur<<<tiles, 32, 0, stream>>>(h0, h1, dat);

    int rows = NBATCH * IMG_H;
    nedt_rowpass<<<(rows + 255) / 256, 256, 0, stream>>>(dat, outp);

    nedt_colpass<<<dim3(IMG_W, NBATCH), 256, 0, stream>>>(outp);
}